// _RoIPooling_12764642803808
// MI455X (gfx1250) — compile-verified
//
#include <hip/hip_runtime.h>
#include <hip/hip_bf16.h>
#include <stdint.h>

// ---------------- problem constants (match reference setup_inputs) ----------
constexpr int   Bc    = 4;      // batch
constexpr int   Cc    = 256;    // channels
constexpr int   Hc    = 50;     // feature height
constexpr int   Wc    = 50;     // feature width
constexpr int   PHc   = 7;      // pooled height
constexpr int   PWc   = 7;      // pooled width
constexpr float SC    = 0.0625f;
constexpr int   CPB   = 4;      // channels staged per block -> <= 40 KB LDS
constexpr int   TDMC  = 2;      // channels staged by the TDM (0..TDMC-1)
constexpr int   BLOCK = 256;    // 8 wave32s

typedef unsigned int u32x4 __attribute__((ext_vector_type(4)));
typedef int          i32x8 __attribute__((ext_vector_type(8)));
typedef int          i32x4 __attribute__((ext_vector_type(4)));

// gfx1250 async global->LDS copy (ASYNCcnt): per-lane (ldsAddr, globalAddr),
// one B128 instruction moves 512 B per wave.
__device__ __forceinline__ void async_g2lds_b128(uint32_t lds_byte_addr,
                                                 const float* gaddr) {
  asm volatile("global_load_async_to_lds_b128 %0, %1, off"
               :: "v"(lds_byte_addr), "v"(gaddr)
               : "memory");
}
__device__ __forceinline__ void wait_async0() {
  asm volatile("s_wait_asynccnt 0" ::: "memory");
}

__global__ __launch_bounds__(BLOCK) void roi_pool_kernel(
    const float* __restrict__ feat,   // [B,C,H,W]
    const float* __restrict__ rois,   // [N,5]
    float* __restrict__ out)          // [N,C,PH,PW]
{
  __shared__ float s_feat[CPB * Hc * Wc];   // 40 KB

  const int n   = blockIdx.y;        // roi index
  const int c0  = blockIdx.x * CPB;  // first channel of this chunk
  const int tid = threadIdx.x;

  // ---- per-ROI params (uniform across the block) ----
  const float* r = rois + (size_t)n * 5;
  const int   b  = (int)r[0];
  const float x1 = rintf(r[1] * SC);      // RNE == jnp.round
  const float y1 = rintf(r[2] * SC);
  const float x2 = rintf(r[3] * SC);
  const float y2 = rintf(r[4] * SC);
  const float roi_w = fmaxf(x2 - x1 + 1.0f, 1.0f);
  const float roi_h = fmaxf(y2 - y1 + 1.0f, 1.0f);
  const float bin_w = roi_w / (float)PWc;
  const float bin_h = roi_h / (float)PHc;

  // bin row range = [hs[0], he[PH-1]) (monotone in i); round outward to even
  // rows so each staged channel chunk is a multiple of 100 dwords and starts
  // 16-byte aligned in both global memory and LDS.
  const int h0 = (int)fminf(fmaxf(y1, 0.0f), (float)Hc);
  const int h1 = (int)fminf(fmaxf(ceilf((float)PHc * bin_h) + y1, 0.0f), (float)Hc);
  const int h0e    = h0 & ~1;            // even, <= h0
  const int h1e    = (h1 + 1) & ~1;      // even, >= h1 (H = 50 is even)
  const int nrowsE = h1e - h0e;          // even row count
  const int chunk  = nrowsE * Wc;        // dwords per channel, multiple of 4

  const float* gbase = feat + (((size_t)b * Cc + c0) * Hc + h0e) * Wc;

  if (h1 > h0) {
    const int wave = tid >> 5;
    if (wave == 0) {
      // ---- wave 0: Tensor Data Mover stages channels [0, TDMC) as one 3D
      //      tile (50 x nrowsE x TDMC), strides 50 / 2500. EXEC-ignored
      //      per-wave instruction -> issue from one wave only.
      const uint64_t gaddr = (uint64_t)gbase;
      const uint32_t lds   = (uint32_t)(size_t)(&s_feat[0]);

      u32x4 g0;
      g0[0] = 1u;                                   // count=1 (valid), user mode
      g0[1] = lds;                                  // lds_addr (bytes)
      g0[2] = (uint32_t)gaddr;                      // global_addr[31:0]
      g0[3] = (uint32_t)((gaddr >> 32) & 0x1FFFFFFu) | (2u << 30); // [56:32]|type=2

      i32x8 g1;
      g1[0] = (2 << 16);                            // data_size=2 -> 4 bytes
      g1[1] = (Wc & 0xFFFF) << 16;                  // tensor_dim0[15:0] = 50
      g1[2] = (nrowsE & 0xFFFF) << 16;              // dim0 hi=0 | tensor_dim1 lo
      g1[3] = (Wc << 16);                           // dim1 hi=0 | tile_dim0 = 50
      g1[4] = (TDMC << 16) | (nrowsE & 0xFFFF);     // tile_dim2 | tile_dim1
      g1[5] = Wc;                                   // tensor_dim0_stride = 50
      g1[6] = (Hc * Wc) << 16;                      // dim1_stride lo = 2500
      g1[7] = 0;                                    // tensor_dim1_stride hi

      i32x4 g2 = { TDMC, 0, 0, 0 };                 // tensor_dim2 (z length)
      i32x4 g3 = { 0, 0, 0, 0 };
      i32x8 gz = { 0, 0, 0, 0, 0, 0, 0, 0 };        // unused slot (NULL desc)

      __builtin_amdgcn_tensor_load_to_lds(g0, g1, g2, g3, gz, 0);
      __builtin_amdgcn_s_wait_tensorcnt(0);         // DMA done -> LDS valid
    } else {
      // ---- waves 1..7: async B128 copies stage channels [TDMC, CPB)
      //      concurrently with the TDM DMA.
      const int lane = tid - 32;                    // 0..223 across 7 waves
      for (int cl = TDMC; cl < CPB; ++cl) {
        const float* g = gbase + (size_t)cl * (Hc * Wc);
        float*       s = s_feat + cl * chunk;
        for (int t = lane * 4; t < chunk; t += (BLOCK - 32) * 4) {
          async_g2lds_b128((uint32_t)(size_t)(&s[t]), g + t);
        }
      }
      wait_async0();                                // this wave's copies done
    }
  }
  __syncthreads();                                  // publish stage to all waves

  // ---- compute CPB*7*7 = 196 bins from LDS ----
  if (tid < CPB * PHc * PWc) {
    const int cl = tid / (PHc * PWc);
    const int rr = tid % (PHc * PWc);
    const int i  = rr / PWc;
    const int j  = rr % PWc;

    const int hs = (int)fminf(fmaxf(floorf((float)i       * bin_h) + y1, 0.0f), (float)Hc);
    const int he = (int)fminf(fmaxf(ceilf ((float)(i + 1) * bin_h) + y1, 0.0f), (float)Hc);
    const int ws = (int)fminf(fmaxf(floorf((float)j       * bin_w) + x1, 0.0f), (float)Wc);
    const int we = (int)fminf(fmaxf(ceilf ((float)(j + 1) * bin_w) + x1, 0.0f), (float)Wc);

    float m = 0.0f;                          // empty bins -> 0 (matches ref)
    if (he > hs && we > ws) {
      m = -3.402823466e+38f;
      const float* sc = s_feat + (size_t)cl * chunk;
      for (int h = hs; h < he; ++h) {
        const float* row = sc + (h - h0e) * Wc;
        for (int w = ws; w < we; ++w)
          m = fmaxf(m, row[w]);
      }
    }
    out[(((size_t)n * Cc + (c0 + cl)) * PHc + i) * PWc + j] = m;
  }
}

extern "C" void kernel_launch(void* const* d_in, const int* in_sizes, int n_in,
                              void* d_out, int out_size, void* d_ws, size_t ws_size,
                              hipStream_t stream) {
  const float* feat = (const float*)d_in[0];
  const float* rois = (const float*)d_in[1];
  float*       out  = (float*)d_out;
  const int N = in_sizes[1] / 5;           // 256 rois

  dim3 grid(Cc / CPB, N);                  // (64, 256) blocks
  dim3 block(BLOCK);
  roi_pool_kernel<<<grid, block, 0, stream>>>(feat, rois, out);
}